// GCNPROTEINS_30666066493991
// MI455X (gfx1250) — compile-verified
//
#include <hip/hip_runtime.h>

typedef __attribute__((ext_vector_type(2))) float v2f;
typedef __attribute__((ext_vector_type(8))) float v8f;

#define DOUT 128

// ---------------- degree computation ----------------
__global__ void k_init_deg(float* deg, int N) {
  int i = blockIdx.x * blockDim.x + threadIdx.x;
  if (i < N) deg[i] = 1.0f;  // self-loop contributes 1
}

__global__ void k_deg_accum(float* deg, const long long* __restrict__ dst, long long E) {
  long long i = (long long)blockIdx.x * blockDim.x + threadIdx.x;
  if (i < E) unsafeAtomicAdd(&deg[(int)dst[i]], 1.0f);
}

__global__ void k_rsqrt(float* deg, int N) {
  int i = blockIdx.x * blockDim.x + threadIdx.x;
  if (i < N) deg[i] = rsqrtf(deg[i]);  // deg >= 1 always (self-loops)
}

// ---------------- GEMM: out[N x 128] = A[N x K] * W[K x 128] ----------------
// One block = 16 node rows; 8 waves, wave w owns output cols [16w, 16w+16).
// K-loop steps of 4 using V_WMMA_F32_16X16X4_F32 (ISA 7.12.2 f32 layouts).
__global__ __launch_bounds__(256) void k_gemm_wmma(const float* __restrict__ A,
                                                   const float* __restrict__ W,
                                                   float* __restrict__ out,
                                                   int N, int K) {
  __shared__ float sA[16 * 128];
  int row0 = blockIdx.x * 16;
  int nld = 16 * K;
  for (int i = threadIdx.x; i < nld; i += 256) {
    int r = i / K, c = i - r * K;
    int gr = row0 + r;
    if (gr >= N) gr = N - 1;  // clamp (reads only; tail rows masked on store)
    sA[i] = A[(long long)gr * K + c];
  }
  __syncthreads();

  int lane = threadIdx.x & 31;
  int wave = threadIdx.x >> 5;
  int col0 = wave * 16;
  int half = lane >> 4;   // 0: K pair {k0,k0+1}, 1: {k0+2,k0+3}
  int l15  = lane & 15;   // A: row index; B/D: column index

  v8f acc = {};
  for (int k0 = 0; k0 < K; k0 += 4) {
    int kk = k0 + half * 2;
    v2f a, b;
    a.x = sA[l15 * K + kk];
    a.y = sA[l15 * K + kk + 1];
    b.x = W[kk * DOUT + col0 + l15];
    b.y = W[(kk + 1) * DOUT + col0 + l15];
    acc = __builtin_amdgcn_wmma_f32_16x16x4_f32(false, a, false, b,
                                                (short)0, acc, false, false);
  }
#pragma unroll
  for (int v = 0; v < 8; ++v) {
    int r = row0 + half * 8 + v;  // D: VGPR v -> M = v (lanes 0-15) / v+8 (16-31)
    if (r < N) out[(long long)r * DOUT + col0 + l15] = acc[v];
  }
}

// ---------------- aggregation ----------------
// agg[i,:] = bias + dinv[i]^2 * tmp[i,:]   (self-loop term + bias)
__global__ void k_agg_init(float* __restrict__ agg, const float* __restrict__ tmp,
                           const float* __restrict__ dinv, const float* __restrict__ bias,
                           int N) {
  long long i = (long long)blockIdx.x * blockDim.x + threadIdx.x;
  long long total = (long long)N * 32;
  if (i >= total) return;
  int node = (int)(i >> 5);
  int q = ((int)i & 31) * 4;
  float di = dinv[node];
  float nrm = di * di;
  float4 t  = *(const float4*)(tmp + (long long)node * DOUT + q);
  float4 bq = *(const float4*)(bias + q);
  float4 r;
  r.x = bq.x + nrm * t.x;
  r.y = bq.y + nrm * t.y;
  r.z = bq.z + nrm * t.z;
  r.w = bq.w + nrm * t.w;
  *(float4*)(agg + (long long)node * DOUT + q) = r;
}

// agg[dst,:] += dinv[src]*dinv[dst] * tmp[src,:]  — 32 threads (float4 each) per edge
__global__ void k_edge_scatter(float* __restrict__ agg, const float* __restrict__ tmp,
                               const float* __restrict__ dinv,
                               const long long* __restrict__ src,
                               const long long* __restrict__ dst,
                               long long E) {
  long long i = (long long)blockIdx.x * blockDim.x + threadIdx.x;
  long long total = E * 32;
  if (i >= total) return;
  long long e = i >> 5;          // uniform per wave -> scalarized loads
  int q = ((int)i & 31) * 4;
  int s = (int)src[e];
  int d = (int)dst[e];
  float nrm = dinv[s] * dinv[d];
  float4 t = *(const float4*)(tmp + (long long)s * DOUT + q);
  float* ap = agg + (long long)d * DOUT + q;
  unsafeAtomicAdd(ap + 0, nrm * t.x);
  unsafeAtomicAdd(ap + 1, nrm * t.y);
  unsafeAtomicAdd(ap + 2, nrm * t.z);
  unsafeAtomicAdd(ap + 3, nrm * t.w);
}

// ---------------- epilogue: out = [relu](agg [+ resid]) ----------------
__global__ void k_epilogue(const float* agg, const float* resid, float* out,
                           long long n4, int do_relu, int do_resid) {
  long long i = (long long)blockIdx.x * blockDim.x + threadIdx.x;
  if (i >= n4) return;
  float4 v = ((const float4*)agg)[i];
  if (do_resid) {
    float4 r = ((const float4*)resid)[i];
    v.x += r.x; v.y += r.y; v.z += r.z; v.w += r.w;
  }
  if (do_relu) {
    v.x = fmaxf(v.x, 0.0f); v.y = fmaxf(v.y, 0.0f);
    v.z = fmaxf(v.z, 0.0f); v.w = fmaxf(v.w, 0.0f);
  }
  ((float4*)out)[i] = v;
}

extern "C" void kernel_launch(void* const* d_in, const int* in_sizes, int n_in,
                              void* d_out, int out_size, void* d_ws, size_t ws_size,
                              hipStream_t stream) {
  const float*     x  = (const float*)d_in[0];
  const long long* ei = (const long long*)d_in[1];   // [2, E] int64, row-major
  const float*     W1 = (const float*)d_in[2];
  const float*     b1 = (const float*)d_in[3];
  const float*     W2 = (const float*)d_in[4];
  const float*     b2 = (const float*)d_in[5];
  const float*     W3 = (const float*)d_in[6];
  const float*     b3 = (const float*)d_in[7];

  const int FIN = 8;
  int N = in_sizes[0] / FIN;
  long long E = in_sizes[1] / 2;
  const long long* srcI = ei;
  const long long* dstI = ei + E;

  float* out = (float*)d_out;
  float* ws  = (float*)d_ws;

  // workspace layout (floats): deg/dinv [N] | tmp [N*128] | h [N*128]
  float* deg = ws;
  size_t off = ((size_t)N + 1023) & ~(size_t)1023;
  float* tmp = ws + off;
  float* h   = tmp + (size_t)N * DOUT;
  float* agg = out;  // d_out doubles as the aggregation buffer

  const int t = 256;
  int gN = (N + t - 1) / t;
  int gE = (int)((E + t - 1) / t);

  long long nodeWork = (long long)N * 32;
  long long edgeWork = E * 32;
  long long n4 = (long long)N * DOUT / 4;
  int gNode = (int)((nodeWork + t - 1) / t);
  int gEdge = (int)((edgeWork + t - 1) / t);
  int g4    = (int)((n4 + t - 1) / t);
  int gGemm = (N + 15) / 16;

  // ---- normalization: dinv = rsqrt(1 + in-degree) ----
  k_init_deg <<<gN, t, 0, stream>>>(deg, N);
  k_deg_accum<<<gE, t, 0, stream>>>(deg, dstI, E);
  k_rsqrt    <<<gN, t, 0, stream>>>(deg, N);
  float* dinv = deg;

  // ---- layer 1: h = relu(scatter(norm * (x@W1)) + b1) ----
  k_gemm_wmma   <<<gGemm, 256, 0, stream>>>(x, W1, tmp, N, FIN);
  k_agg_init    <<<gNode, t, 0, stream>>>(agg, tmp, dinv, b1, N);
  k_edge_scatter<<<gEdge, t, 0, stream>>>(agg, tmp, dinv, srcI, dstI, E);
  k_epilogue    <<<g4,    t, 0, stream>>>(agg, nullptr, h, n4, 1, 0);

  // ---- layer 2: h = relu(gcn(h,W2,b2) + h) ----
  k_gemm_wmma   <<<gGemm, 256, 0, stream>>>(h, W2, tmp, N, DOUT);
  k_agg_init    <<<gNode, t, 0, stream>>>(agg, tmp, dinv, b2, N);
  k_edge_scatter<<<gEdge, t, 0, stream>>>(agg, tmp, dinv, srcI, dstI, E);
  k_epilogue    <<<g4,    t, 0, stream>>>(agg, h, h, n4, 1, 1);

  // ---- layer 3: out = gcn(h,W3,b3) + h ----
  k_gemm_wmma   <<<gGemm, 256, 0, stream>>>(h, W3, tmp, N, DOUT);
  k_agg_init    <<<gNode, t, 0, stream>>>(agg, tmp, dinv, b3, N);
  k_edge_scatter<<<gEdge, t, 0, stream>>>(agg, tmp, dinv, srcI, dstI, E);
  k_epilogue    <<<g4,    t, 0, stream>>>(agg, h, out, n4, 0, 1);
}